// RiemannianLayerNormV2_91001767068239
// MI455X (gfx1250) — compile-verified
//
#include <hip/hip_runtime.h>
#include <stdint.h>

// RiemannianLayerNorm, B=8 L=4096 D=1024 (row length 1025).
// Bandwidth-bound (AI ~2.5 flop/B): target = HBM roof (~268MB / 23.3TB/s ~= 11.5us).
// CDNA5 paths: async global->LDS DMA in GVS (saddr) mode with split ASYNCcnt
// waits for load/compute overlap + wave32 shuffle reductions.

#define DD      1024
#define ROWLEN  1025
#define NT      256
#define NW      8       // 256 threads / wave32

__device__ __forceinline__ float sanitize_f(float v) {
  // jnp.nan_to_num(nan=0, posinf=1e4, neginf=-1e4)
  if (v != v) return 0.0f;
  if (v ==  __builtin_inff()) return  1.0e4f;
  if (v == -__builtin_inff()) return -1.0e4f;
  return v;
}

// CDNA5 async DMA, GVS mode: one shared voff VGPR + SGPR base; the 24-bit
// instruction offset advances BOTH the LDS and global addresses (ISA 08 §4.4).
#define ASYNC_LD_B32(lds_addr, voff, sbase, imm)                         \
  asm volatile("global_load_async_to_lds_b32 %0, %1, %2 offset:" #imm   \
               :: "v"(lds_addr), "v"(voff), "s"(sbase) : "memory")

#define WAIT_ASYNC(n) asm volatile("s_wait_asynccnt " #n ::: "memory")

// Sum-reduce two floats across the 256-thread block (wave32 shuffle tree
// + 8-partial LDS combine; every thread ends with the block sums).
__device__ __forceinline__ void block_reduce2(float& a, float& b, volatile float* sred) {
  #pragma unroll
  for (int off = 16; off > 0; off >>= 1) {
    a += __shfl_down(a, off, 32);
    b += __shfl_down(b, off, 32);
  }
  const int wave = threadIdx.x >> 5;
  const int lane = threadIdx.x & 31;
  if (lane == 0) { sred[wave] = a; sred[wave + NW] = b; }
  __syncthreads();
  a = 0.0f; b = 0.0f;
  #pragma unroll
  for (int w = 0; w < NW; ++w) { a += sred[w]; b += sred[w + NW]; }
}

extern "C" __global__ void __launch_bounds__(NT, 1)
rln_rows(const float* __restrict__ x, const float* __restrict__ gamma,
         const float* __restrict__ beta, float* __restrict__ out) {
  __shared__ float s_x[DD];
  __shared__ float s_g[DD];
  __shared__ float s_b[DD];
  __shared__ float s_red1[2 * NW];
  __shared__ float s_red2[2 * NW];
  __shared__ float s_x0;

  const int    tid  = threadIdx.x;
  const size_t base = (size_t)blockIdx.x * (size_t)ROWLEN;
  const float* __restrict__ xrow = x + base + 1;

  // LDS byte addresses (generic shared ptr low 32 bits = LDS offset).
  const uint32_t voff = (uint32_t)tid * 4u;
  const uint32_t lx   = (uint32_t)(uintptr_t)(void*)s_x + voff;
  const uint32_t lg   = (uint32_t)(uintptr_t)(void*)s_g + voff;
  const uint32_t lb   = (uint32_t)(uintptr_t)(void*)s_b + voff;

  // ---- Stage row first (needed for pass 1), then gamma/beta (pass 2) ----
  ASYNC_LD_B32(lx, voff, xrow, 0);
  ASYNC_LD_B32(lx, voff, xrow, 1024);
  ASYNC_LD_B32(lx, voff, xrow, 2048);
  ASYNC_LD_B32(lx, voff, xrow, 3072);
  ASYNC_LD_B32(lg, voff, gamma, 0);
  ASYNC_LD_B32(lg, voff, gamma, 1024);
  ASYNC_LD_B32(lg, voff, gamma, 2048);
  ASYNC_LD_B32(lg, voff, gamma, 3072);
  ASYNC_LD_B32(lb, voff, beta, 0);
  ASYNC_LD_B32(lb, voff, beta, 1024);
  ASYNC_LD_B32(lb, voff, beta, 2048);
  ASYNC_LD_B32(lb, voff, beta, 3072);
  if (tid == 0) s_x0 = x[base];   // x0 scalar (visible after reduce barrier)

  // Async loads complete in order: cnt<=8 means the 4 x-loads landed.
  // Pass 1 reads only this thread's own staged values -> no barrier needed.
  WAIT_ASYNC(8);

  // ---- Pass 1: S1 = sum(xs), S2 = sum(xs^2)  (v = c*xs => mu,var analytic) ----
  float xv[4];
  float s1 = 0.0f, s2 = 0.0f;
  #pragma unroll
  for (int k = 0; k < 4; ++k) {
    const float v = sanitize_f(s_x[tid + k * NT]);
    xv[k] = v;
    s1 += v;
    s2 += v * v;
  }
  block_reduce2(s1, s2, s_red1);   // contains __syncthreads (covers s_x0 too)

  const float x0   = sanitize_f(s_x0);
  const float x0c  = fmaxf(x0, 1.0f + 1e-7f);
  const float dist = __logf(x0c + sqrtf(x0c * x0c - 1.0f));   // arccosh
  const float xsn  = fmaxf(sqrtf(s2), 1e-7f);
  const float c    = dist / xsn;
  const float invd = 1.0f / (float)DD;
  const float mu   = c * s1 * invd;
  const float var  = fmaxf(c * c * s2 * invd - mu * mu, 0.0f);
  const float rstd = __frsqrt_rn(var + 1e-5f);

  // ---- Pass 2: v_norm = clip((c*x - mu)*rstd*g + b, +-5); n2 = sum(v_norm^2) ----
  WAIT_ASYNC(0);                   // gamma/beta landed (own values only)
  float vn[4];
  float s3 = 0.0f, zero = 0.0f;
  #pragma unroll
  for (int k = 0; k < 4; ++k) {
    const int ch = tid + k * NT;
    float t = (c * xv[k] - mu) * rstd * s_g[ch] + s_b[ch];
    t = fminf(fmaxf(t, -5.0f), 5.0f);
    vn[k] = t;
    s3 += t * t;
  }
  block_reduce2(s3, zero, s_red2);

  const float n2    = s3;
  const float n     = sqrtf(n2);
  const float n_c   = fminf(fmaxf(n, 1e-7f), 1.5f);
  const float e     = __expf(n_c);
  const float sh    = 0.5f * (e - 1.0f / e);                 // sinh(n_c)
  const float scale = sh / fmaxf(n, 1e-7f);

  // ---- Emit: xs_out = scale*v_norm ; x0_out = sqrt(1 + scale^2 * n2) ----
  float* __restrict__ orow = out + base;
  #pragma unroll
  for (int k = 0; k < 4; ++k) {
    const int ch = tid + k * NT;
    __builtin_nontemporal_store(scale * vn[k], orow + 1 + ch);
  }
  if (tid == 0) {
    __builtin_nontemporal_store(sqrtf(1.0f + scale * scale * n2), orow);
  }
}

extern "C" void kernel_launch(void* const* d_in, const int* in_sizes, int n_in,
                              void* d_out, int out_size, void* d_ws, size_t ws_size,
                              hipStream_t stream) {
  const float* x     = (const float*)d_in[0];
  const float* gamma = (const float*)d_in[1];
  const float* beta  = (const float*)d_in[2];
  float*       out   = (float*)d_out;
  const int nrows = out_size / ROWLEN;   // 8*4096 = 32768
  rln_rows<<<dim3(nrows), dim3(NT), 0, stream>>>(x, gamma, beta, out);
}